// ADCActivation_52106543235511
// MI455X (gfx1250) — compile-verified
//
#include <hip/hip_runtime.h>
#include <stdint.h>

// ADC activation: out = 2*(searchsorted(thr, x, right)/256 - 0.5)
// HBM-bound op (512 MB total traffic ~22us @ 23.3 TB/s). Strategy:
//  1) tiny setup kernel builds an exact 2048-bin LUT (base_count, bin_count)
//  2) main kernel: non-temporal b128 streaming loads/stores, 1 LDS LUT hit +
//     rare sorted-prefix fixup per element; LDS staged via gfx1250 async
//     global->LDS loads (ASYNCcnt path).

#define NB      2048      // LUT bins (8 KB)
#define MAX_THR 2048      // max thresholds held in LDS

// native vector type: required by __builtin_nontemporal_load/store
typedef float v4f __attribute__((ext_vector_type(4)));

// Shared binning helper: MUST be bit-identical between build and lookup.
// (v - t0) * invw cannot be fp-contracted (sub feeding mul is not an fma
// pattern), so plain ops are deterministic. fmaxf first absorbs any NaN
// from the degenerate 0*inf case into bin 0, identically in both kernels.
// f is monotone non-decreasing in v, so bin(t) < bin(x) => t <= x and
// bin(t) > bin(x) => t > x; same-bin thresholds are compared explicitly,
// making the LUT method exact (bit-identical counts to the reference).
__device__ __forceinline__ int bin_of(float v, float t0, float invw) {
  float f = (v - t0) * invw;
  f = fminf(fmaxf(f, 0.0f), (float)(NB - 1));
  return (int)f;  // truncation == floor for f >= 0
}

__device__ __forceinline__ float inv_width(float t0, float tlast) {
  float d = tlast - t0;
  return (d > 0.0f) ? ((float)NB / d) : 0.0f;
}

__device__ __forceinline__ float adc_from_lut(float v, float t0, float invw,
                                              const uint32_t* __restrict__ sLUT,
                                              const float* __restrict__ sThr) {
  int b = bin_of(v, t0, invw);
  uint32_t e = sLUT[b];
  int c = (int)(e >> 16);          // thresholds strictly below this bin
  int k = (int)(e & 0xFFFFu);      // thresholds inside this bin
  int s = c;
  for (int j = 0; j < k; ++j) {    // expected ~0.12 iterations; bin slice is
    if (sThr[s + j] > v) break;    // sorted, so first ">" ends the count
    ++c;
  }
  return __fmaf_rn((float)c, 0.0078125f, -1.0f);  // c/128 - 1, exact
}

// ---------------- setup: build LUT in global scratch ----------------
__global__ void __launch_bounds__(256)
build_lut_kernel(const float* __restrict__ thr, int m, uint32_t* __restrict__ lut) {
  __shared__ float sThr[MAX_THR];
  for (int i = threadIdx.x; i < m; i += blockDim.x) sThr[i] = thr[i];
  __syncthreads();
  const float t0 = sThr[0];
  const float invw = inv_width(t0, sThr[m - 1]);
  for (int b = threadIdx.x; b < NB; b += blockDim.x) {
    int start = 0, cnt = 0;
    for (int i = 0; i < m; ++i) {
      int bi = bin_of(sThr[i], t0, invw);
      start += (bi < b) ? 1 : 0;
      cnt   += (bi == b) ? 1 : 0;
    }
    lut[b] = ((uint32_t)start << 16) | (uint32_t)cnt;
  }
}

// ---------------- main streaming kernel ----------------
__global__ void __launch_bounds__(256)
adc_lut_kernel(const float* __restrict__ x, const float* __restrict__ thr, int m,
               const uint32_t* __restrict__ lut, float* __restrict__ out,
               long long n) {
  __shared__ __align__(16) uint32_t sLUT[NB];
  __shared__ __align__(16) float    sThr[MAX_THR];
  const int tid = threadIdx.x;

#if defined(__gfx1250__)
  {
    // Stage 2048-dword LUT: each of 256 threads pulls 8 dwords as 2x async b128.
    unsigned long long g0 = (unsigned long long)(const void*)(lut + tid * 8);
    unsigned int l0 = (unsigned int)(size_t)&sLUT[tid * 8];
    asm volatile("global_load_async_to_lds_b128 %0, %1, off"
                 :: "v"(l0), "v"(g0) : "memory");
    asm volatile("global_load_async_to_lds_b128 %0, %1, off offset:16"
                 :: "v"(l0), "v"(g0) : "memory");
    // Stage thresholds (m <= MAX_THR) one dword per lane.
    for (int i = tid; i < m; i += 256) {
      unsigned long long gt = (unsigned long long)(const void*)(thr + i);
      unsigned int lt = (unsigned int)(size_t)&sThr[i];
      asm volatile("global_load_async_to_lds_b32 %0, %1, off"
                   :: "v"(lt), "v"(gt) : "memory");
    }
    asm volatile("s_wait_asynccnt 0x0" ::: "memory");
  }
#else
  for (int i = tid; i < NB; i += 256) sLUT[i] = lut[i];
  for (int i = tid; i < m;  i += 256) sThr[i] = thr[i];
#endif
  __syncthreads();

  const float t0 = sThr[0];
  const float invw = inv_width(t0, sThr[m - 1]);

  const long long nvec   = n >> 2;
  const long long stride = (long long)gridDim.x * blockDim.x;
  const v4f* __restrict__ x4 = (const v4f*)x;
  v4f* __restrict__ o4 = (v4f*)out;

  for (long long i = (long long)blockIdx.x * blockDim.x + tid; i < nvec; i += stride) {
    // one-shot streaming data: non-temporal so the hot LUT stays cached
    v4f v = __builtin_nontemporal_load(&x4[i]);
    v4f r;
    r.x = adc_from_lut(v.x, t0, invw, sLUT, sThr);
    r.y = adc_from_lut(v.y, t0, invw, sLUT, sThr);
    r.z = adc_from_lut(v.z, t0, invw, sLUT, sThr);
    r.w = adc_from_lut(v.w, t0, invw, sLUT, sThr);
    __builtin_nontemporal_store(r, &o4[i]);
  }
  // tail (n % 4 elements)
  for (long long i = (nvec << 2) + (long long)blockIdx.x * blockDim.x + tid;
       i < n; i += stride) {
    float r = adc_from_lut(x[i], t0, invw, sLUT, sThr);
    __builtin_nontemporal_store(r, &out[i]);
  }
}

// ---------------- fallback: LDS binary search (ws too small / m too big) ----
__global__ void __launch_bounds__(256)
adc_bs_kernel(const float* __restrict__ x, const float* __restrict__ thr, int m,
              float* __restrict__ out, long long n) {
  __shared__ float sThr[MAX_THR];
  const bool useL = (m <= MAX_THR);
  if (useL) {
    for (int i = threadIdx.x; i < m; i += blockDim.x) sThr[i] = thr[i];
    __syncthreads();
  }
  const float* __restrict__ T = useL ? (const float*)sThr : thr;
  const long long stride = (long long)gridDim.x * blockDim.x;
  for (long long i = (long long)blockIdx.x * blockDim.x + threadIdx.x; i < n;
       i += stride) {
    float v = x[i];
    int lo = 0, hi = m;
    while (lo < hi) {
      int mid = (lo + hi) >> 1;
      if (T[mid] <= v) lo = mid + 1; else hi = mid;
    }
    out[i] = __fmaf_rn((float)lo, 0.0078125f, -1.0f);
  }
}

extern "C" void kernel_launch(void* const* d_in, const int* in_sizes, int n_in,
                              void* d_out, int out_size, void* d_ws, size_t ws_size,
                              hipStream_t stream) {
  const float* x   = (const float*)d_in[0];
  const float* thr = (const float*)d_in[1];
  const long long n = (long long)in_sizes[0];
  const int m = in_sizes[1];
  float* out = (float*)d_out;

  long long nthreads_needed = (n + 3) >> 2;
  int blocks = (int)((nthreads_needed + 255) / 256);
  if (blocks > 8192) blocks = 8192;  // grid-stride; amortizes 9KB LDS staging
  if (blocks < 1) blocks = 1;

  if (m >= 1 && m <= MAX_THR && ws_size >= (size_t)NB * sizeof(uint32_t)) {
    uint32_t* lut = (uint32_t*)d_ws;
    build_lut_kernel<<<1, 256, 0, stream>>>(thr, m, lut);
    adc_lut_kernel<<<blocks, 256, 0, stream>>>(x, thr, m, lut, out, n);
  } else {
    adc_bs_kernel<<<blocks, 256, 0, stream>>>(x, thr, m, out, n);
  }
}